// FlowPlusPlusLayer_16810501997233
// MI455X (gfx1250) — compile-verified
//
#include <hip/hip_runtime.h>
#include <math.h>

// ---------------- problem constants (fixed by reference) ----------------
#define DIMQ   16
#define HQ     512
#define BQ     16384
#define KQ     4
#define ROWS   32            // rows per made workgroup
#define TPB    256           // threads per made workgroup (8 wave32s)
#define H0STR  (HQ + 2)      // padded LDS row stride (bank-conflict avoidance)
#define XSSTR  (DIMQ + 2)

typedef float v2f __attribute__((ext_vector_type(2)));
typedef float v8f __attribute__((ext_vector_type(8)));

static __device__ __forceinline__ v8f wmma4(v2f a, v2f b, v8f c) {
  // D = A(16x4,f32) * B(4x16,f32) + C(16x16,f32)   -> v_wmma_f32_16x16x4_f32
  return __builtin_amdgcn_wmma_f32_16x16x4_f32(false, a, false, b, (short)0, c,
                                               false, false);
}

static __device__ __forceinline__ float leaky(float v) {
  return v >= 0.f ? v : 0.01f * v;
}
static __device__ __forceinline__ float softplusf(float v) {
  return v > 0.f ? v + log1pf(__expf(-v)) : log1pf(__expf(v));
}

// gathered output-column id for step t, local col n in [0,14)
static __device__ __forceinline__ int colid(int t, int n) {
  if (n == 0) return t;                 // log_a
  if (n == 1) return 16 + t;            // b
  if (n < 6)  return 32  + t + 16 * (n - 2);   // raw_pi (K=4)
  if (n < 10) return 96  + t + 16 * (n - 6);   // mu
  return 160 + t + 16 * (n - 10);              // log_sigma
}

// ---------------- setup kernels ----------------

__global__ void init_out_kernel(float* __restrict__ out, int n) {
  int i = blockIdx.x * blockDim.x + threadIdx.x;
  if (i < n) out[i] = 0.f;
}

// Pack masked W0 (16 x 512) into WMMA-B fragment order:
// Wp0[nt][ks][lane][j],  b.x/b.y = W[4ks+2h+j][16nt+lm],  mask M0: (c%15) >= k
__global__ void pack_w0_kernel(const float* __restrict__ W0, float* __restrict__ Wp0) {
  int gid  = blockIdx.x * blockDim.x + threadIdx.x;     // 32nt * 4ks * 32lane = 4096
  int lane = gid & 31;
  int ks   = (gid >> 5) & 3;
  int nt   = gid >> 7;
  int h = lane >> 4, lm = lane & 15;
  int c  = nt * 16 + lm;
  int k0 = 4 * ks + 2 * h;
  #pragma unroll
  for (int j = 0; j < 2; ++j) {
    int k = k0 + j;
    float w = ((c % 15) >= k) ? W0[k * HQ + c] : 0.f;
    Wp0[gid * 2 + j] = w;
  }
}

// Pack masked W1 (512 x 512); mask M1: (c%15) >= (k%15)
__global__ void pack_w1_kernel(const float* __restrict__ W1, float* __restrict__ Wp1) {
  int gid  = blockIdx.x * blockDim.x + threadIdx.x;     // 32nt * 128ks * 32 = 131072
  int lane = gid & 31;
  int ks   = (gid >> 5) & 127;
  int nt   = gid >> 12;
  int h = lane >> 4, lm = lane & 15;
  int c  = nt * 16 + lm;
  int k0 = 4 * ks + 2 * h;
  #pragma unroll
  for (int j = 0; j < 2; ++j) {
    int k = k0 + j;
    float w = ((c % 15) >= (k % 15)) ? W1[k * HQ + c] : 0.f;
    Wp1[gid * 2 + j] = w;
  }
}

// Pack, per t, the 14 needed masked W2 columns (512 x 16, cols 14..15 zero).
// mask M2 for these columns reduces to: t > (k % 15)
__global__ void pack_w2_kernel(const float* __restrict__ W2, float* __restrict__ Wp2) {
  int gid  = blockIdx.x * blockDim.x + threadIdx.x;     // 16t * 128ks * 32 = 65536
  int lane = gid & 31;
  int ks   = (gid >> 5) & 127;
  int t    = gid >> 12;
  int h = lane >> 4, lm = lane & 15;
  int k0 = 4 * ks + 2 * h;
  #pragma unroll
  for (int j = 0; j < 2; ++j) {
    int k = k0 + j;
    float w = 0.f;
    if (lm < 14 && (t > (k % 15))) w = W2[k * 224 + colid(t, lm)];
    Wp2[gid * 2 + j] = w;
  }
}

__global__ void pack_b2_kernel(const float* __restrict__ b2, float* __restrict__ b2g) {
  int i = threadIdx.x;          // 256 = 16t * 16n
  int t = i >> 4, n = i & 15;
  b2g[i] = (n < 14) ? b2[colid(t, n)] : 0.f;
}

// ---------------- fused MADE kernel (one autoregressive step) ----------------
// grid = B/ROWS = 512 blocks, 256 threads; dynamic LDS holds x, h0, h1, partials.
__global__ void made_kernel(const float* __restrict__ xin,
                            const float* __restrict__ Wp0, const float* __restrict__ b0,
                            const float* __restrict__ Wp1, const float* __restrict__ b1,
                            const float* __restrict__ Wp2, const float* __restrict__ b2g,
                            const int* __restrict__ order, int iter,
                            float* __restrict__ out14) {
  extern __shared__ float sm[];
  float* xs  = sm;                                   // [32][18]
  float* h0s = xs  + ROWS * XSSTR;                   // [32][514]
  float* h1s = h0s + ROWS * H0STR;                   // [32][514]
  float* red = h1s + ROWS * H0STR;                   // [8][16][16]

  const int tid  = threadIdx.x;
  const int lane = tid & 31;
  const int wid  = tid >> 5;
  const int h    = lane >> 4;       // which K-half of the fragment this lane holds
  const int lm   = lane & 15;
  const int row0 = blockIdx.x * ROWS;
  const int t    = order[iter];

  // ---- stage x tile (32 x 16) into LDS ----
  {
    int f = tid * 2;
    int r = f >> 4, c = f & 15;
    xs[r * XSSTR + c]     = xin[(row0 + r) * DIMQ + c];
    xs[r * XSSTR + c + 1] = xin[(row0 + r) * DIMQ + c + 1];
  }
  __syncthreads();

  // ---- layer 0: h0 = leaky(x @ W0m + b0), K = 16 ----
  {
    v8f z = {0.f, 0.f, 0.f, 0.f, 0.f, 0.f, 0.f, 0.f};
    v8f acc[2][4];
    #pragma unroll
    for (int m = 0; m < 2; ++m)
      #pragma unroll
      for (int j = 0; j < 4; ++j) acc[m][j] = z;

    #pragma unroll
    for (int ks = 0; ks < 4; ++ks) {
      v2f a0 = *(const v2f*)&xs[lm * XSSTR + 4 * ks + 2 * h];
      v2f a1 = *(const v2f*)&xs[(16 + lm) * XSSTR + 4 * ks + 2 * h];
      #pragma unroll
      for (int j = 0; j < 4; ++j) {
        v2f b = *(const v2f*)&Wp0[(((wid * 4 + j) * 4 + ks) * 32 + lane) * 2];
        acc[0][j] = wmma4(a0, b, acc[0][j]);
        acc[1][j] = wmma4(a1, b, acc[1][j]);
      }
    }
    #pragma unroll
    for (int j = 0; j < 4; ++j) {
      int c = (wid * 4 + j) * 16 + lm;
      float bias = b0[c];
      #pragma unroll
      for (int m = 0; m < 2; ++m)
        #pragma unroll
        for (int v = 0; v < 8; ++v)
          h0s[(m * 16 + v + 8 * h) * H0STR + c] = leaky(acc[m][j][v] + bias);
    }
  }
  __syncthreads();

  // ---- layer 1: h1 = leaky(h0 @ W1m + b1), K = 512 ----
  {
    v8f z = {0.f, 0.f, 0.f, 0.f, 0.f, 0.f, 0.f, 0.f};
    v8f acc[2][4];
    #pragma unroll
    for (int m = 0; m < 2; ++m)
      #pragma unroll
      for (int j = 0; j < 4; ++j) acc[m][j] = z;

    #pragma unroll 4
    for (int ks = 0; ks < 128; ++ks) {
      v2f a0 = *(const v2f*)&h0s[lm * H0STR + 4 * ks + 2 * h];
      v2f a1 = *(const v2f*)&h0s[(16 + lm) * H0STR + 4 * ks + 2 * h];
      #pragma unroll
      for (int j = 0; j < 4; ++j) {
        v2f b = *(const v2f*)&Wp1[(((wid * 4 + j) * 128 + ks) * 32 + lane) * 2];
        acc[0][j] = wmma4(a0, b, acc[0][j]);
        acc[1][j] = wmma4(a1, b, acc[1][j]);
      }
    }
    #pragma unroll
    for (int j = 0; j < 4; ++j) {
      int c = (wid * 4 + j) * 16 + lm;
      float bias = b1[c];
      #pragma unroll
      for (int m = 0; m < 2; ++m)
        #pragma unroll
        for (int v = 0; v < 8; ++v)
          h1s[(m * 16 + v + 8 * h) * H0STR + c] = leaky(acc[m][j][v] + bias);
    }
  }
  __syncthreads();

  // ---- layer 2 (trimmed): out14 = h1 @ W2g[t] + b2g[t], K split over waves ----
  {
    int mt = wid & 1;          // which 16-row M tile
    int q  = wid >> 1;         // K-quarter
    v8f acc = {0.f, 0.f, 0.f, 0.f, 0.f, 0.f, 0.f, 0.f};
    #pragma unroll 4
    for (int ks = q * 32; ks < q * 32 + 32; ++ks) {
      v2f a = *(const v2f*)&h1s[(mt * 16 + lm) * H0STR + 4 * ks + 2 * h];
      v2f b = *(const v2f*)&Wp2[((t * 128 + ks) * 32 + lane) * 2];
      acc = wmma4(a, b, acc);
    }
    #pragma unroll
    for (int v = 0; v < 8; ++v)
      red[wid * 256 + (v + 8 * h) * 16 + lm] = acc[v];
  }
  __syncthreads();

  // reduce the 4 K-partials per M tile, add bias, write out14
  for (int e = tid; e < ROWS * 16; e += TPB) {
    int r = e >> 4, c = e & 15;
    int mt = r >> 4, r16 = r & 15;
    float s = b2g[t * 16 + c];
    #pragma unroll
    for (int q = 0; q < 4; ++q) s += red[(2 * q + mt) * 256 + r16 * 16 + c];
    out14[(row0 + r) * 16 + c] = s;
  }
}

// ---------------- per-row flow / bisection kernel ----------------
__global__ void flow_kernel(const float* __restrict__ u,
                            const float* __restrict__ out14,
                            const int* __restrict__ order, int iter,
                            float* __restrict__ xout, float* __restrict__ logd) {
  int row = blockIdx.x * blockDim.x + threadIdx.x;
  int t = order[iter];
  const float* o = out14 + row * 16;

  float la = 0.1f * tanhf(o[0]);
  float bb = o[1];
  float rp[KQ], mu[KQ], ls[KQ];
  #pragma unroll
  for (int k = 0; k < KQ; ++k) {
    rp[k] = o[2 + k];
    mu[k] = o[6 + k];
    ls[k] = o[10 + k];
  }
  // log_softmax over mixture logits
  float mx = fmaxf(fmaxf(rp[0], rp[1]), fmaxf(rp[2], rp[3]));
  float se = 0.f;
  #pragma unroll
  for (int k = 0; k < KQ; ++k) se += __expf(rp[k] - mx);
  float lse = mx + __logf(se);
  float lp[KQ];
  #pragma unroll
  for (int k = 0; k < KQ; ++k) lp[k] = rp[k] - lse;

  float uu = u[row * DIMQ + t];
  float s  = (uu - bb) * __expf(-la);
  // log(y) + log1p(-y) = -softplus(-s) - softplus(s)
  float logd_t = -la - softplusf(-s) - softplusf(s);
  float y = 1.f / (1.f + __expf(-s));

  float invs[KQ], pi[KQ];
  #pragma unroll
  for (int k = 0; k < KQ; ++k) { invs[k] = __expf(-ls[k]); pi[k] = __expf(lp[k]); }

  // bisection: CDF(mid) = sum_k pi_k * sigmoid(invs_k * (mid - mu_k))
  float lo = -1000.f, hi = 1000.f;
  for (int it2 = 0; it2 < 50; ++it2) {
    float mid = 0.5f * (lo + hi);
    float val = 0.f;
    #pragma unroll
    for (int k = 0; k < KQ; ++k)
      val += __fdividef(pi[k], 1.f + __expf(-invs[k] * (mid - mu[k])));
    lo = (val < y) ? mid : lo;
    hi = (val > y) ? mid : hi;
  }
  float xt = 0.5f * (lo + hi);

  // mixture log-pdf at xt
  float lpdf[KQ];
  #pragma unroll
  for (int k = 0; k < KQ; ++k) {
    float z = invs[k] * (xt - mu[k]);
    lpdf[k] = z - ls[k] - 2.f * softplusf(z) + lp[k];
  }
  float m2 = fmaxf(fmaxf(lpdf[0], lpdf[1]), fmaxf(lpdf[2], lpdf[3]));
  float s2 = 0.f;
  #pragma unroll
  for (int k = 0; k < KQ; ++k) s2 += __expf(lpdf[k] - m2);
  logd_t -= (m2 + __logf(s2));

  xout[row * DIMQ + t] = xt;
  logd[row] += logd_t;
}

// ---------------- launch ----------------
extern "C" void kernel_launch(void* const* d_in, const int* in_sizes, int n_in,
                              void* d_out, int out_size, void* d_ws, size_t ws_size,
                              hipStream_t stream) {
  const float* u  = (const float*)d_in[0];
  const float* W0 = (const float*)d_in[1];
  const float* b0 = (const float*)d_in[2];
  const float* W1 = (const float*)d_in[3];
  const float* b1 = (const float*)d_in[4];
  const float* W2 = (const float*)d_in[5];
  const float* b2 = (const float*)d_in[6];
  // d_in[7..9] = masks (computed analytically in pack kernels)
  const int* order = (const int*)d_in[10];

  float* ws    = (float*)d_ws;
  float* Wp0   = ws;                  // 32*4*64      =   8192 floats
  float* Wp1   = Wp0 + 8192;          // 32*128*64    = 262144
  float* Wp2   = Wp1 + 262144;        // 16*128*64    = 131072
  float* b2g   = Wp2 + 131072;        // 16*16        =    256
  float* out14 = b2g + 256;           // B*16         = 262144

  float* xout = (float*)d_out;        // B x 16
  float* logd = xout + BQ * DIMQ;     // B

  // one-time (per launch) setup — deterministic, graph-capturable
  init_out_kernel<<<(out_size + 255) / 256, 256, 0, stream>>>((float*)d_out, out_size);
  pack_w0_kernel<<<16, 256, 0, stream>>>(W0, Wp0);
  pack_w1_kernel<<<512, 256, 0, stream>>>(W1, Wp1);
  pack_w2_kernel<<<256, 256, 0, stream>>>(W2, Wp2);
  pack_b2_kernel<<<1, 256, 0, stream>>>(b2, b2g);

  const size_t lds_bytes =
      (size_t)(ROWS * XSSTR + 2 * ROWS * H0STR + 8 * 256) * sizeof(float); // ~139 KB

  for (int i = 0; i < DIMQ; ++i) {
    made_kernel<<<BQ / ROWS, TPB, lds_bytes, stream>>>(
        xout, Wp0, b0, Wp1, b1, Wp2, b2g, order, i, out14);
    flow_kernel<<<BQ / 256, 256, 0, stream>>>(u, out14, order, i, xout, logd);
  }
}